// ChannelAttention_Deform_LKA_sequential_13417477833352
// MI455X (gfx1250) — compile-verified
//
#include <hip/hip_runtime.h>
#include <math.h>

// ---------------- CDNA5 WMMA (fp32, 16x16x4) ----------------
typedef float v2f __attribute__((ext_vector_type(2)));
typedef float v8f __attribute__((ext_vector_type(8)));

#define WMMA_F32_4(a, b, c) \
  __builtin_amdgcn_wmma_f32_16x16x4_f32(false, (a), false, (b), (short)0, (c), false, false)

static constexpr int N_ = 32768;   // 32*32*32 spatial
// B=2, C=64, HEADS=4, hd=16, spatial strides: dim0=1024, dim1=32, dim2=1

// =====================================================================
// K1: row-major GEMM  OUT[m,j] = sum_k A[m*K+k] * W[j*K+k] (+bias[j])
//     one wave -> one 16x16 tile, f32 WMMA, K stepped by 4
// =====================================================================
__global__ void k_gemm_rm(const float* __restrict__ A, const float* __restrict__ W,
                          const float* __restrict__ bias, float* __restrict__ OUT,
                          int M, int K, int J) {
  int wave = (int)((blockIdx.x * blockDim.x + threadIdx.x) >> 5);
  int lane = threadIdx.x & 31;
  int jt = J >> 4;
  int mt_i = wave / jt, jt_i = wave % jt;
  if (mt_i * 16 >= M) return;                 // wave-uniform guard
  int m0 = mt_i * 16, j0 = jt_i * 16;
  int lm = lane & 15, sel = lane >> 4;
  v8f acc = {};
  const float* arow = A + (size_t)(m0 + lm) * K;
  const float* wrow = W + (size_t)(j0 + lm) * K;
  for (int k0 = 0; k0 < K; k0 += 4) {
    v2f a, b;
    a.x = arow[k0 + sel * 2 + 0];
    a.y = arow[k0 + sel * 2 + 1];
    b.x = wrow[k0 + sel * 2 + 0];
    b.y = wrow[k0 + sel * 2 + 1];
    acc = WMMA_F32_4(a, b, acc);
  }
  float bj = bias ? bias[j0 + lm] : 0.0f;
  for (int g = 0; g < 8; ++g) {
    int m = m0 + g + sel * 8;
    OUT[(size_t)m * J + j0 + lm] = acc[g] + bj;
  }
}

// =====================================================================
// K2: split qkv, L2-normalize q,k over N, transpose to [B*h*hd, N]
//     grid.x = 128 rows (b*64 + h*16 + c), grid.y = 3 (q,k,v)
// =====================================================================
__global__ void k_qkv_split(const float* __restrict__ qkv, float* __restrict__ qn,
                            float* __restrict__ kn, float* __restrict__ vt) {
  int row  = blockIdx.x;          // b*64 + h*16 + c
  int part = blockIdx.y;          // 0=q 1=k 2=v
  int b = row >> 6;
  int hc = row & 63;
  const float* src = qkv + (size_t)b * N_ * 192 + part * 64 + hc;
  float* dst = (part == 0 ? qn : part == 1 ? kn : vt) + (size_t)row * N_;
  __shared__ float red[256];
  float scale = 1.0f;
  if (part < 2) {
    float ss = 0.f;
    for (int n = threadIdx.x; n < N_; n += 256) {
      float v = src[(size_t)n * 192];
      ss += v * v;
    }
    red[threadIdx.x] = ss;
    __syncthreads();
    for (int s = 128; s > 0; s >>= 1) {
      if ((int)threadIdx.x < s) red[threadIdx.x] += red[threadIdx.x + s];
      __syncthreads();
    }
    scale = 1.0f / fmaxf(sqrtf(red[0]), 1e-12f);
  }
  for (int n = threadIdx.x; n < N_; n += 256)
    dst[n] = src[(size_t)n * 192] * scale;
}

// =====================================================================
// K3: attn[b,h,c,d] = softmax_d( temp[h] * sum_n qn[c,n]*kn[d,n] )
//     8 waves split K=32768, f32 WMMA, LDS cross-wave reduce + softmax
// =====================================================================
__global__ void k_attn(const float* __restrict__ qn, const float* __restrict__ kn,
                       const float* __restrict__ temp, float* __restrict__ attn) {
  int bh = blockIdx.x;  // 0..7
  int wave = threadIdx.x >> 5, lane = threadIdx.x & 31;
  int lm = lane & 15, sel = lane >> 4;
  const float* q = qn + (size_t)bh * 16 * N_;
  const float* k = kn + (size_t)bh * 16 * N_;
  v8f acc = {};
  for (int ch = wave; ch < N_ / 4; ch += 8) {
    int n0 = ch * 4;
    v2f a, b;
    const float* qr = q + (size_t)lm * N_ + n0 + sel * 2;
    const float* kr = k + (size_t)lm * N_ + n0 + sel * 2;
    a.x = qr[0]; a.y = qr[1];
    b.x = kr[0]; b.y = kr[1];
    acc = WMMA_F32_4(a, b, acc);
  }
  __shared__ float sm[8 * 256];
  for (int g = 0; g < 8; ++g) {
    int m = g + sel * 8, d = lm;
    sm[wave * 256 + m * 16 + d] = acc[g];
  }
  __syncthreads();
  __shared__ float s2[256];
  int t = threadIdx.x;
  float s = 0.f;
  for (int w = 0; w < 8; ++w) s += sm[w * 256 + t];
  s *= temp[bh & 3];
  s2[t] = s;
  __syncthreads();
  int mrow = t >> 4;
  float mx = -INFINITY;
  for (int d = 0; d < 16; ++d) mx = fmaxf(mx, s2[mrow * 16 + d]);
  float sum = 0.f;
  for (int d = 0; d < 16; ++d) sum += __expf(s2[mrow * 16 + d] - mx);
  attn[bh * 256 + t] = __expf(s - mx) / sum;
}

// =====================================================================
// K4: x_ca[b,n,h*16+c] = sum_d attn[bh,c,d] * v[bh,d,n]
// =====================================================================
__global__ void k_av(const float* __restrict__ attn, const float* __restrict__ vt,
                     float* __restrict__ xca) {
  int bh = blockIdx.y;
  int n = blockIdx.x * 256 + threadIdx.x;
  __shared__ float at[256];
  at[threadIdx.x] = attn[bh * 256 + threadIdx.x];
  __syncthreads();
  int b = bh >> 2, h = bh & 3;
  float vv[16];
  for (int d = 0; d < 16; ++d) vv[d] = vt[((size_t)bh * 16 + d) * N_ + n];
  float* dst = xca + ((size_t)b * N_ + n) * 64 + h * 16;
  for (int c = 0; c < 16; ++c) {
    float a0 = 0.f;
    for (int d = 0; d < 16; ++d) a0 += at[c * 16 + d] * vv[d];
    dst[c] = a0;
  }
}

// =====================================================================
// K5: LayerNorm over C=64 of [B,N,C] input, write transposed [B,C,N]
// =====================================================================
__global__ void k_ln1(const float* __restrict__ in, const float* __restrict__ w,
                      const float* __restrict__ bb, float* __restrict__ out) {
  size_t p = (size_t)blockIdx.x * 256 + threadIdx.x;   // b*N + n
  size_t b = p >> 15;
  size_t n = p & (N_ - 1);
  float v[64];
  float mean = 0.f;
  for (int c = 0; c < 64; ++c) { v[c] = in[p * 64 + c]; mean += v[c]; }
  mean *= 0.015625f;
  float var = 0.f;
  for (int c = 0; c < 64; ++c) { float d = v[c] - mean; var += d * d; }
  var *= 0.015625f;
  float inv = rsqrtf(var + 1e-5f);
  for (int c = 0; c < 64; ++c)
    out[((b * 64 + c) << 15) + n] = (v[c] - mean) * inv * w[c] + bb[c];
}

// =====================================================================
// K6: channel GEMM on [B,C,N]:  OUT[b,o,n] = sum_c IN[b,c,n]*W[o,c] + bias
//     fused epilogues: mode 1 = exact GELU; 2 = extra*val (gate);
//                      3 = val + extra (shortcut)
// =====================================================================
__global__ void k_gemm_cn(const float* __restrict__ IN, const float* __restrict__ W,
                          const float* __restrict__ bias, float* __restrict__ OUT,
                          const float* __restrict__ extra, int mode) {
  int wave_g = (int)((blockIdx.x * blockDim.x + threadIdx.x) >> 5);
  int lane = threadIdx.x & 31, lm = lane & 15, sel = lane >> 4;
  int nt = wave_g & 2047;
  int rest = wave_g >> 11;
  int ot = rest & 3, b = rest >> 2;
  int o0 = ot * 16, n0 = nt * 16;
  v8f acc = {};
  const float* wrow = W + (size_t)(o0 + lm) * 64;
  const float* bcol = IN + (size_t)b * 64 * N_ + n0 + lm;
  for (int k0 = 0; k0 < 64; k0 += 4) {
    v2f a, bv;
    int kk = k0 + sel * 2;
    a.x = wrow[kk]; a.y = wrow[kk + 1];
    bv.x = bcol[(size_t)kk * N_];
    bv.y = bcol[(size_t)(kk + 1) * N_];
    acc = WMMA_F32_4(a, bv, acc);
  }
  for (int g = 0; g < 8; ++g) {
    int o = o0 + g + sel * 8;
    size_t idx = ((size_t)b * 64 + o) * N_ + n0 + lm;
    float v = acc[g] + bias[o];
    if (mode == 1)      v = 0.5f * v * (1.0f + erff(v * 0.70710678118654752f));
    else if (mode == 2) v = extra[idx] * v;
    else if (mode == 3) v = v + extra[idx];
    OUT[idx] = v;
  }
}

// =====================================================================
// K7: depthwise 3D conv (ks^3 taps, pad, dilation), layout [B,C,N]
// =====================================================================
__global__ void k_dw(const float* __restrict__ IN, const float* __restrict__ W,
                     const float* __restrict__ bias, float* __restrict__ OUT,
                     int ks, int pad, int dil) {
  size_t gid = (size_t)blockIdx.x * 256 + threadIdx.x;  // (b*64+c)*N + n
  int n = (int)(gid & (N_ - 1));
  int bc = (int)(gid >> 15);
  int c = bc & 63;
  int z = n & 31, y = (n >> 5) & 31, x = n >> 10;
  const float* src = IN + (size_t)bc * N_;
  int taps = ks * ks * ks;
  const float* wt = W + (size_t)c * taps;
  float acc = bias[c];
  for (int t = 0; t < taps; ++t) {
    int tx = t / (ks * ks), ty = (t / ks) % ks, tz = t % ks;
    int ix = x + tx * dil - pad, iy = y + ty * dil - pad, iz = z + tz * dil - pad;
    if ((unsigned)ix < 32u && (unsigned)iy < 32u && (unsigned)iz < 32u)
      acc += src[ix * 1024 + iy * 32 + iz] * wt[t];
  }
  OUT[gid] = acc;
}

// =====================================================================
// K8: offset conv (3^3, 64->81 ch) as implicit GEMM with WMMA.
//     M padded 81 -> 96 (6 o-tiles), zero-fill invalid rows/taps.
// =====================================================================
__global__ void k_offconv(const float* __restrict__ IN, const float* __restrict__ W,
                          const float* __restrict__ bias, float* __restrict__ OUT) {
  int wave_g = (int)((blockIdx.x * blockDim.x + threadIdx.x) >> 5);
  int lane = threadIdx.x & 31, lm = lane & 15, sel = lane >> 4;
  int nt = wave_g & 2047;
  int rest = wave_g >> 11;
  int ot = rest % 6, b = rest / 6;
  int o0 = ot * 16, n0 = nt * 16;
  int n = n0 + lm;
  int z = n & 31, y = (n >> 5) & 31, x = n >> 10;
  int oA = o0 + lm;
  bool oval = oA < 81;
  const float* inb = IN + (size_t)b * 64 * N_;
  v8f acc = {};
  for (int t = 0; t < 27; ++t) {
    int dx = t / 9 - 1, dy = (t / 3) % 3 - 1, dz = t % 3 - 1;
    int ix = x + dx, iy = y + dy, iz = z + dz;
    bool valid = (unsigned)ix < 32u && (unsigned)iy < 32u && (unsigned)iz < 32u;
    int cx = min(max(ix, 0), 31), cy = min(max(iy, 0), 31), cz = min(max(iz, 0), 31);
    size_t sidx = (size_t)(cx * 1024 + cy * 32 + cz);
    for (int k0 = 0; k0 < 64; k0 += 4) {
      v2f a, bv;
      int kk = k0 + sel * 2;
      float a0 = oval ? W[((size_t)oA * 64 + kk) * 27 + t] : 0.f;
      float a1 = oval ? W[((size_t)oA * 64 + kk + 1) * 27 + t] : 0.f;
      float b0 = inb[(size_t)kk * N_ + sidx];
      float b1 = inb[(size_t)(kk + 1) * N_ + sidx];
      a.x = a0; a.y = a1;
      bv.x = valid ? b0 : 0.f;
      bv.y = valid ? b1 : 0.f;
      acc = WMMA_F32_4(a, bv, acc);
    }
  }
  for (int g = 0; g < 8; ++g) {
    int o = o0 + g + sel * 8;
    if (o < 81)
      OUT[((size_t)b * 81 + o) * N_ + n0 + lm] = acc[g] + bias[o];
  }
}

// =====================================================================
// K9: deformable conv. One block = (b, 16-position tile), 4 waves.
//     Per tap: 16 thr compute trilinear corners -> 128 thr gather
//     samp[64x16] to LDS -> each wave runs 16 f32-WMMAs (its o-tile).
// =====================================================================
__global__ void k_dcn(const float* __restrict__ X, const float* __restrict__ OFF,
                      const float* __restrict__ Wd, const float* __restrict__ bias,
                      float* __restrict__ OUT) {
  int b  = blockIdx.x >> 11;      // N/16 = 2048 tiles per batch
  int nt = blockIdx.x & 2047;
  int n0 = nt * 16;
  int tid = threadIdx.x;
  int wave = tid >> 5, lane = tid & 31, lm = lane & 15, sel = lane >> 4;
  __shared__ float samp[64 * 16];
  __shared__ float wgt[8][16];
  __shared__ int   widx[8][16];
  v8f acc = {};
  const float* xb = X + (size_t)b * 64 * N_;
  for (int k = 0; k < 27; ++k) {
    if (tid < 16) {
      int n = n0 + tid;
      int kz = k / 9, ky = (k / 3) % 3, kx = k % 3;
      float pz = (float)(n >> 10)       + (float)(kz - 1) + OFF[((size_t)b * 81 + k * 3 + 0) * N_ + n];
      float py = (float)((n >> 5) & 31) + (float)(ky - 1) + OFF[((size_t)b * 81 + k * 3 + 1) * N_ + n];
      float px = (float)(n & 31)        + (float)(kx - 1) + OFF[((size_t)b * 81 + k * 3 + 2) * N_ + n];
      float z0f = floorf(pz), y0f = floorf(py), x0f = floorf(px);
      float wz = pz - z0f, wy = py - y0f, wx = px - x0f;
      int z0 = (int)z0f, y0 = (int)y0f, x0 = (int)x0f;
      for (int j = 0; j < 8; ++j) {
        int dz = j >> 2, dy = (j >> 1) & 1, dx = j & 1;
        int iz = z0 + dz, iy = y0 + dy, ix = x0 + dx;
        bool valid = (unsigned)iz < 32u && (unsigned)iy < 32u && (unsigned)ix < 32u;
        float cw = (dz ? wz : 1.f - wz) * (dy ? wy : 1.f - wy) * (dx ? wx : 1.f - wx);
        int cz = min(max(iz, 0), 31), cy = min(max(iy, 0), 31), cx = min(max(ix, 0), 31);
        wgt[j][tid]  = valid ? cw : 0.f;
        widx[j][tid] = cz * 1024 + cy * 32 + cx;
      }
    }
    __syncthreads();
    for (int cell = tid; cell < 1024; cell += 128) {
      int c = cell >> 4, nl = cell & 15;
      const float* xc = xb + (size_t)c * N_;
      float v = 0.f;
      for (int j = 0; j < 8; ++j) v += wgt[j][nl] * xc[widx[j][nl]];
      samp[c * 16 + nl] = v;
    }
    __syncthreads();
    int o0 = wave * 16;
    for (int k0 = 0; k0 < 64; k0 += 4) {
      v2f a, bv;
      int kk = k0 + sel * 2;
      a.x = Wd[((size_t)(o0 + lm) * 64 + kk) * 27 + k];
      a.y = Wd[((size_t)(o0 + lm) * 64 + kk + 1) * 27 + k];
      bv.x = samp[kk * 16 + lm];
      bv.y = samp[(kk + 1) * 16 + lm];
      acc = WMMA_F32_4(a, bv, acc);
    }
    __syncthreads();
  }
  for (int g = 0; g < 8; ++g) {
    int o = wave * 16 + g + sel * 8;
    OUT[((size_t)b * 64 + o) * N_ + n0 + lm] = acc[g] + bias[o];
  }
}

// =====================================================================
// K10: faithful "reshape-after-permute" + LayerNorm:
//  T2[b,n2,c2] = y[b, n2&63, c2*512 + (n2>>6)]; LN over c2; out [B,N,C]
// =====================================================================
__global__ void k_ln2v(const float* __restrict__ in, const float* __restrict__ w,
                       const float* __restrict__ bb, float* __restrict__ out) {
  size_t p = (size_t)blockIdx.x * 256 + threadIdx.x;  // b*N + n2
  size_t b = p >> 15;
  int n2 = (int)(p & (N_ - 1));
  int ch = n2 & 63;
  int base = n2 >> 6;
  const float* src = in + ((b * 64 + ch) << 15) + base;
  float v[64];
  float mean = 0.f;
  for (int c2 = 0; c2 < 64; ++c2) { v[c2] = src[(size_t)c2 * 512]; mean += v[c2]; }
  mean *= 0.015625f;
  float var = 0.f;
  for (int c2 = 0; c2 < 64; ++c2) { float d = v[c2] - mean; var += d * d; }
  var *= 0.015625f;
  float inv = rsqrtf(var + 1e-5f);
  for (int c2 = 0; c2 < 64; ++c2)
    out[p * 64 + c2] = (v[c2] - mean) * inv * w[c2] + bb[c2];
}

// =====================================================================
extern "C" void kernel_launch(void* const* d_in, const int* in_sizes, int n_in,
                              void* d_out, int out_size, void* d_ws, size_t ws_size,
                              hipStream_t stream) {
  (void)in_sizes; (void)n_in; (void)out_size; (void)ws_size;
  const float* x        = (const float*)d_in[0];
  const float* temp     = (const float*)d_in[1];
  const float* qkv_w    = (const float*)d_in[2];
  const float* norm_w   = (const float*)d_in[3];
  const float* norm_b   = (const float*)d_in[4];
  const float* proj1_w  = (const float*)d_in[5];
  const float* proj1_b  = (const float*)d_in[6];
  const float* conv0_w  = (const float*)d_in[7];
  const float* conv0_b  = (const float*)d_in[8];
  const float* convsp_w = (const float*)d_in[9];
  const float* convsp_b = (const float*)d_in[10];
  const float* off_w    = (const float*)d_in[11];
  const float* off_b    = (const float*)d_in[12];
  const float* dcn_w    = (const float*)d_in[13];
  const float* dcn_b    = (const float*)d_in[14];
  const float* conv1_w  = (const float*)d_in[15];
  const float* conv1_b  = (const float*)d_in[16];
  const float* proj2_w  = (const float*)d_in[17];
  const float* proj2_b  = (const float*)d_in[18];
  const float* norm2_w  = (const float*)d_in[19];
  const float* norm2_b  = (const float*)d_in[20];
  const float* out_w    = (const float*)d_in[21];
  const float* out_b    = (const float*)d_in[22];

  float* ws = (float*)d_ws;
  // Phase A
  float* QKV  = ws;                     // 12,582,912  [B,N,192]
  float* QN   = ws + 12582912;          //  4,194,304  [B*4*16, N]
  float* KN   = QN + 4194304;
  float* VT   = KN + 4194304;
  float* ATTN = VT + 4194304;           //      2,048
  float* XCA  = ATTN + 2048;            //  4,194,304  [B,N,64]
  // Phase B (reuses Phase-A space; XCA dead after ln1)
  float* X5   = ws;                     //  4,194,304  [B,64,N] (shortcut)
  float* U    = ws + 4194304;
  float* A0   = ws + 8388608;
  float* A1   = ws + 12582912;
  float* OFFS = ws + 16777216;          //  5,308,416  [B,81,N]
  float* DCN  = ws + 22085632;
  float* Y    = ws + 26279936;
  float* Y2   = ws + 30474240;
  float* XS   = ws + 34668544;          // end 38,862,848 floats (~148 MiB)

  // 1. qkv projection: [65536,64] x [192,64]^T
  k_gemm_rm<<<(4096 * 12) / 8, 256, 0, stream>>>(x, qkv_w, nullptr, QKV, 65536, 64, 192);
  // 2. split + L2 normalize q,k; transpose v
  dim3 g2(128, 3);
  k_qkv_split<<<g2, 256, 0, stream>>>(QKV, QN, KN, VT);
  // 3. channel attention scores + softmax
  k_attn<<<8, 256, 0, stream>>>(QN, KN, temp, ATTN);
  // 4. attn @ v -> x_ca [B,N,C]
  dim3 g4(N_ / 256, 8);
  k_av<<<g4, 256, 0, stream>>>(ATTN, VT, XCA);
  // 5. LayerNorm -> X5 [B,C,N] (also shortcut)
  k_ln1<<<(2 * N_) / 256, 256, 0, stream>>>(XCA, norm_w, norm_b, X5);
  // 6. proj1 + exact GELU -> U
  k_gemm_cn<<<(2 * 4 * 2048) / 8, 256, 0, stream>>>(X5, proj1_w, proj1_b, U, nullptr, 1);
  // 7. depthwise 5^3 pad 2 -> A0
  k_dw<<<(2 * 64 * N_) / 256, 256, 0, stream>>>(U, conv0_w, conv0_b, A0, 5, 2, 1);
  // 8. depthwise 7^3 dil 3 pad 9 -> A1
  k_dw<<<(2 * 64 * N_) / 256, 256, 0, stream>>>(A0, convsp_w, convsp_b, A1, 7, 9, 3);
  // 9. offset conv (implicit GEMM) -> OFFS [B,81,N]
  k_offconv<<<(2 * 6 * 2048) / 8, 256, 0, stream>>>(A1, off_w, off_b, OFFS);
  // 10. deformable conv -> DCN
  k_dcn<<<2 * 2048, 128, 0, stream>>>(A1, OFFS, dcn_w, dcn_b, DCN);
  // 11. conv1, gated by U -> Y
  k_gemm_cn<<<(2 * 4 * 2048) / 8, 256, 0, stream>>>(DCN, conv1_w, conv1_b, Y, U, 2);
  // 12. proj2 + shortcut -> Y2
  k_gemm_cn<<<(2 * 4 * 2048) / 8, 256, 0, stream>>>(Y, proj2_w, proj2_b, Y2, X5, 3);
  // 13. faithful permute-reshape view + LayerNorm -> XS [B,N,C]
  k_ln2v<<<(2 * N_) / 256, 256, 0, stream>>>(Y2, norm2_w, norm2_b, XS);
  // 14. output projection -> d_out [B,N,64]
  k_gemm_rm<<<(4096 * 4) / 8, 256, 0, stream>>>(XS, out_w, out_b, (float*)d_out, 65536, 64, 64);
}